// Model_51058571215596
// MI455X (gfx1250) — compile-verified
//
#include <hip/hip_runtime.h>

// Problem constants from the reference: B=4, S=4096, D=4096, fp32.
#define BB 4
#define SS 4096
#define DD 4096
#define ROWS (BB * SS)          // 16384 rows
#define EPSILON 1e-6f

typedef __attribute__((ext_vector_type(2))) float v2f;
typedef __attribute__((ext_vector_type(8))) float v8f;

// ---------------------------------------------------------------------------
// Kernel 0: zero the global abs-max accumulator (bit-pattern max on uint,
// valid because every candidate value is >= 0).
// ---------------------------------------------------------------------------
__global__ void gmax_init_kernel(unsigned int* gbits) { *gbits = 0u; }

// ---------------------------------------------------------------------------
// Kernel 1: one 256-thread block per row.
//   - stream x, residual, gamma as float4 (global_load_b128, coalesced)
//   - per-thread: sumsq of (x+r) and max(|x+r| * gamma)
//   - sum-of-squares block reduction via V_WMMA_F32_16X16X4_F32:
//       4 chained WMMAs with B = ones reduce 256 partials on the matrix pipe
//   - abs-max block reduction via wave32 xor-shuffles + LDS
//   - lane 0: rstd -> ws, atomicMax(global_bits, rstd * rowmax)
// ---------------------------------------------------------------------------
__global__ void __launch_bounds__(256) rowstat_kernel(
    const float* __restrict__ x, const float* __restrict__ r,
    const float* __restrict__ gamma, float* __restrict__ rstd_out,
    unsigned int* __restrict__ gbits)
{
    __shared__ float lds_sq[256];
    __shared__ float lds_mx[256];

    const int row = blockIdx.x;
    const int t   = threadIdx.x;
    const size_t base = (size_t)row * DD;

    const float4* __restrict__ x4 = (const float4*)(x + base);
    const float4* __restrict__ r4 = (const float4*)(r + base);
    const float4* __restrict__ g4 = (const float4*)gamma;

    float ss = 0.0f;   // sum of squares partial
    float mx = 0.0f;   // max(|x+r|*gamma) partial (values >= 0)

#pragma unroll
    for (int i = 0; i < 4; ++i) {
        const int c = i * 256 + t;               // float4 index within row
        __builtin_prefetch(x4 + c + 256, 0, 0);  // global_prefetch_b8 (speculative)
        __builtin_prefetch(r4 + c + 256, 0, 0);
        float4 a = x4[c];
        float4 b = r4[c];
        float4 g = g4[c];
        float vx = a.x + b.x, vy = a.y + b.y, vz = a.z + b.z, vw = a.w + b.w;
        ss = fmaf(vx, vx, ss); ss = fmaf(vy, vy, ss);
        ss = fmaf(vz, vz, ss); ss = fmaf(vw, vw, ss);
        mx = fmaxf(mx, fabsf(vx) * g.x);
        mx = fmaxf(mx, fabsf(vy) * g.y);
        mx = fmaxf(mx, fabsf(vz) * g.z);
        mx = fmaxf(mx, fabsf(vw) * g.w);
    }

    lds_sq[t] = ss;
    lds_mx[t] = mx;
    __syncthreads();

    if (t < 32) {                     // exactly wave 0 (wave32): EXEC all ones
        const int lane = t;
        const int rsel = lane & 15;            // A-matrix row M
        const int ksel = (lane >> 4) << 1;     // lanes 0-15 -> K0/K1, 16-31 -> K2/K3

        // Reduce 256 partials: 4 chained f32 WMMAs, A = 16x4 tile of partials,
        // B = ones(4x16).  D[i][j] = sum_k A[i][k] accumulated over 4 tiles.
        v2f bones; bones.x = 1.0f; bones.y = 1.0f;
        v8f acc = {};
#pragma unroll
        for (int cblk = 0; cblk < 4; ++cblk) {
            const int idx = cblk * 64 + rsel * 4 + ksel;
            v2f a;
            a.x = lds_sq[idx];
            a.y = lds_sq[idx + 1];
            acc = __builtin_amdgcn_wmma_f32_16x16x4_f32(
                /*neg_a=*/false, a, /*neg_b=*/false, bones,
                /*c_mod=*/(short)0, acc, /*reuse_a=*/false, /*reuse_b=*/false);
        }
        // C/D layout: VGPR v, lane l -> row M = v + 8*(l>>4).  Sum the 8 local
        // rows then fold the M=8..15 half with one xor-16 shuffle: every lane
        // now holds the full row sum-of-squares.
        float local = acc[0] + acc[1] + acc[2] + acc[3] +
                      acc[4] + acc[5] + acc[6] + acc[7];
        float sumsq = local + __shfl_xor(local, 16, 32);

        // abs-max reduction: 8 LDS values per lane, then 5 xor-shuffle steps.
        float m = 0.0f;
#pragma unroll
        for (int j = 0; j < 8; ++j) m = fmaxf(m, lds_mx[lane * 8 + j]);
#pragma unroll
        for (int off = 16; off > 0; off >>= 1)
            m = fmaxf(m, __shfl_xor(m, off, 32));

        if (lane == 0) {
            const float rstd = rsqrtf(sumsq * (1.0f / (float)DD) + EPSILON);
            rstd_out[row] = rstd;
            // row's contribution to global abs-max of the normalized output
            atomicMax(gbits, __float_as_uint(m * rstd));
        }
    }
}

// ---------------------------------------------------------------------------
// Kernel 2: re-stream x/residual, apply rstd*gamma, quantize with the global
// scale, write float-typed int8 values.  One block per row, float4 I/O.
// ---------------------------------------------------------------------------
__global__ void __launch_bounds__(256) quant_kernel(
    const float* __restrict__ x, const float* __restrict__ r,
    const float* __restrict__ gamma, const float* __restrict__ rstd_arr,
    const unsigned int* __restrict__ gbits, float* __restrict__ out)
{
    const int row = blockIdx.x;
    const int t   = threadIdx.x;
    const size_t base = (size_t)row * DD;

    const float rstd      = rstd_arr[row];                  // uniform -> scalar
    const float gmax      = __uint_as_float(*gbits);        // uniform -> scalar
    const float inv_scale = 127.0f / gmax;                  // q = rint(y*127/max)

    const float4* __restrict__ x4 = (const float4*)(x + base);
    const float4* __restrict__ r4 = (const float4*)(r + base);
    const float4* __restrict__ g4 = (const float4*)gamma;
    float4* __restrict__ o4 = (float4*)(out + base);

#pragma unroll
    for (int i = 0; i < 4; ++i) {
        const int c = i * 256 + t;
        __builtin_prefetch(x4 + c + 256, 0, 0);
        __builtin_prefetch(r4 + c + 256, 0, 0);
        float4 a = x4[c];
        float4 b = r4[c];
        float4 g = g4[c];
        float4 q;
        q.x = fminf(fmaxf(rintf((a.x + b.x) * rstd * g.x * inv_scale), -128.0f), 127.0f);
        q.y = fminf(fmaxf(rintf((a.y + b.y) * rstd * g.y * inv_scale), -128.0f), 127.0f);
        q.z = fminf(fmaxf(rintf((a.z + b.z) * rstd * g.z * inv_scale), -128.0f), 127.0f);
        q.w = fminf(fmaxf(rintf((a.w + b.w) * rstd * g.w * inv_scale), -128.0f), 127.0f);
        o4[c] = q;
    }
}

// ---------------------------------------------------------------------------
// Host-side launcher.  Workspace layout:
//   ws[0..3]      : uint global-max bits
//   ws + 256 B    : float rstd[ROWS]   (64 KB)
// ---------------------------------------------------------------------------
extern "C" void kernel_launch(void* const* d_in, const int* in_sizes, int n_in,
                              void* d_out, int out_size, void* d_ws, size_t ws_size,
                              hipStream_t stream)
{
    const float* x     = (const float*)d_in[0];
    const float* resid = (const float*)d_in[1];
    const float* gamma = (const float*)d_in[2];
    float* out = (float*)d_out;

    unsigned int* gbits = (unsigned int*)d_ws;
    float* rstd = (float*)((char*)d_ws + 256);

    gmax_init_kernel<<<1, 1, 0, stream>>>(gbits);
    rowstat_kernel<<<ROWS, 256, 0, stream>>>(x, resid, gamma, rstd, gbits);
    quant_kernel<<<ROWS, 256, 0, stream>>>(x, resid, gamma, rstd, gbits, out);
}